// CachedGlmExperts_39874476376636
// MI455X (gfx1250) — compile-verified
//
#include <hip/hip_runtime.h>
#include <hip/hip_bf16.h>

#define T_TOKENS 64
#define HID 2048
#define INTER_DIM 1408
#define N_EXP 32
#define TOPK 4
#define BK 32
#define LDS_STRIDE 36   // 32 cols + 4 DW TDM pad: conflict-free transposed reads
#define TILE_FLOATS (128 * LDS_STRIDE)

typedef __attribute__((ext_vector_type(2))) float v2f;
typedef __attribute__((ext_vector_type(8))) float v8f;
typedef __attribute__((ext_vector_type(4))) unsigned int v4u;
typedef __attribute__((ext_vector_type(8))) int v8i;
typedef __attribute__((ext_vector_type(4))) int v4i;

__device__ __forceinline__ v8f wmma_f32_16x16x4(v2f a, v2f b, v8f c) {
  return __builtin_amdgcn_wmma_f32_16x16x4_f32(false, a, false, b, (short)0, c,
                                               false, false);
}

// ---- TDM: DMA one 128-row x 32-float f32 tile (row stride = strideElems)
// into LDS at byte offset lds_addr, with 4-DWORD padding after every 32 DWORDs
// so the LDS tile lands with a 36-float row stride (bank-conflict-free B reads).
__device__ __forceinline__ void tdm_load_tile_f32(unsigned lds_addr,
                                                  const float* gptr,
                                                  unsigned long long strideElems) {
  const unsigned long long ga = (unsigned long long)(size_t)gptr;
  // Group0: count=1(valid), lds_addr, global_addr[56:0], type=2 ("image")
  v4u g0 = {0x1u, lds_addr, (unsigned)ga,
            (unsigned)((ga >> 32) & 0x01FFFFFFu) | 0x80000000u};
  // Group1: data_size=4B(2), pad_enable, pad_interval=32DW(4), pad_amount=4DW(3)
  const unsigned d0 = (2u << 16) | (1u << 20) | (4u << 22) | (3u << 25);
  const unsigned td0 = 0x40000000u;  // huge tensor dims: tiles never OOB
  const unsigned td1 = 0x40000000u;
  v8i g1 = {(int)d0,
            (int)((td0 & 0xFFFFu) << 16),                    // dim0[15:0]<<16
            (int)((td0 >> 16) | ((td1 & 0xFFFFu) << 16)),    // dim0[31:16]|dim1lo
            (int)((td1 >> 16) | (32u << 16)),                // dim1[31:16]|tile_dim0=32
            (int)128,                                        // tile_dim1=128, tile_dim2=0
            (int)(unsigned)(strideElems & 0xFFFFFFFFu),      // dim0_stride[31:0]
            (int)(unsigned)((strideElems >> 32) & 0xFFFFu),  // dim0_stride[47:32]
            0};
  v4i g2 = {0, 0, 0, 0};
  v4i g3 = {0, 0, 0, 0};
#if defined(__clang_major__) && __clang_major__ >= 23
  v8i g4 = {0, 0, 0, 0, 0, 0, 0, 0};
  __builtin_amdgcn_tensor_load_to_lds(g0, g1, g2, g3, g4, 0);
#else
  __builtin_amdgcn_tensor_load_to_lds(g0, g1, g2, g3, 0);
#endif
}

// ---------------- Phase 1: top-k routing + deterministic expert bucketing ----
__global__ __launch_bounds__(64)
void moe_route_kernel(const float* __restrict__ logits,
                      int* __restrict__ cnt, int* __restrict__ off,
                      int* __restrict__ ptok, int* __restrict__ pairof,
                      float* __restrict__ wk) {
  __shared__ int   s_id[T_TOKENS * TOPK];
  __shared__ float s_w[T_TOKENS * TOPK];
  const int t = threadIdx.x;
  if (t < T_TOKENS) {
    float v[N_EXP];
#pragma unroll
    for (int j = 0; j < N_EXP; ++j) v[j] = logits[t * N_EXP + j];
    unsigned taken = 0u;
    float tv[TOPK]; int ti[TOPK];
    for (int k = 0; k < TOPK; ++k) {
      float best = -__builtin_inff(); int bi = 0;
      for (int j = 0; j < N_EXP; ++j)
        if (!((taken >> j) & 1u) && v[j] > best) { best = v[j]; bi = j; }
      taken |= 1u << bi; tv[k] = best; ti[k] = bi;
    }
    const float m = tv[0];
    float ex[TOPK]; float s = 0.f;
    for (int k = 0; k < TOPK; ++k) { ex[k] = __expf(tv[k] - m); s += ex[k]; }
    const float inv = 1.f / s;
    for (int k = 0; k < TOPK; ++k) {
      s_id[t * TOPK + k] = ti[k];
      s_w[t * TOPK + k]  = ex[k] * inv;
    }
  }
  __syncthreads();
  if (t == 0) {  // serial, deterministic bucketing of the 256 pairs
    int c[N_EXP], o[N_EXP], cur[N_EXP];
    for (int e = 0; e < N_EXP; ++e) c[e] = 0;
    for (int p = 0; p < T_TOKENS * TOPK; ++p) c[s_id[p]]++;
    int acc = 0;
    for (int e = 0; e < N_EXP; ++e) {
      o[e] = acc; cur[e] = 0; cnt[e] = c[e]; off[e] = acc; acc += c[e];
    }
    off[N_EXP] = acc;
    for (int tt = 0; tt < T_TOKENS; ++tt)
      for (int k = 0; k < TOPK; ++k) {
        const int e = s_id[tt * TOPK + k];
        const int p = o[e] + cur[e]++;
        ptok[p] = tt;
        pairof[tt * TOPK + k] = p;
        wk[tt * TOPK + k] = s_w[tt * TOPK + k];
      }
  }
}

// ---------------- Phase 2: gate/up GEMMs + fused SiLU*up ---------------------
__global__ __launch_bounds__(256)
void moe_gateup_kernel(const float* __restrict__ x,
                       const float* __restrict__ wg,
                       const float* __restrict__ wu,
                       const int* __restrict__ cnt,
                       const int* __restrict__ off,
                       const int* __restrict__ ptok,
                       float* __restrict__ mixed) {
  __shared__ float lds_g[2][TILE_FLOATS];   // double-buffered TDM destinations
  __shared__ float lds_u[2][TILE_FLOATS];
  const int e = blockIdx.y;
  const int nTok = cnt[e];
  if (nTok == 0) return;
  const int base = off[e];
  const int i0 = blockIdx.x * 128;
  const int tid = threadIdx.x;
  const int lane = tid & 31;
  const int wave = tid >> 5;
  const int lhalf = (lane < 16) ? 0 : 2;
  const int l15 = lane & 15;
  const int bcol = wave * 16 + l15;

  const size_t wbase = ((size_t)e * INTER_DIM + i0) * HID;
  const unsigned ldsg0 = (unsigned)(size_t)(void*)&lds_g[0][0];
  const unsigned ldsg1 = (unsigned)(size_t)(void*)&lds_g[1][0];
  const unsigned ldsu0 = (unsigned)(size_t)(void*)&lds_u[0][0];
  const unsigned ldsu1 = (unsigned)(size_t)(void*)&lds_u[1][0];

  const int nMT = (nTok + 15) >> 4;
  for (int mt = 0; mt < nMT; ++mt) {
    const int arow = mt * 16 + l15;
    const int tok = ptok[base + ((arow < nTok) ? arow : (nTok - 1))];
    const float* xrow = x + (size_t)tok * HID;

    v8f accg = {0.f, 0.f, 0.f, 0.f, 0.f, 0.f, 0.f, 0.f};
    v8f accu = {0.f, 0.f, 0.f, 0.f, 0.f, 0.f, 0.f, 0.f};

    if (wave == 0) {  // prologue: DMA first tiles into buffer 0
      tdm_load_tile_f32(ldsg0, wg + wbase, HID);
      tdm_load_tile_f32(ldsu0, wu + wbase, HID);
    }
    int cur = 0;
    for (int h0 = 0; h0 < HID; h0 += BK) {
      if (wave == 0) __builtin_amdgcn_s_wait_tensorcnt(0);
      __syncthreads();  // current tiles visible to all waves
      if (wave == 0 && (h0 + BK) < HID) {  // DMA next tiles into other buffer
        tdm_load_tile_f32(cur ? ldsg0 : ldsg1, wg + wbase + h0 + BK, HID);
        tdm_load_tile_f32(cur ? ldsu0 : ldsu1, wu + wbase + h0 + BK, HID);
      }
      const float* bg_t = &lds_g[cur][0];
      const float* bu_t = &lds_u[cur][0];
#pragma unroll
      for (int kk = 0; kk < BK; kk += 4) {
        const int kA = kk + lhalf;
        const float2 af  = *(const float2*)(xrow + h0 + kA);
        const float2 bgf = *(const float2*)(bg_t + bcol * LDS_STRIDE + kA);
        const float2 buf = *(const float2*)(bu_t + bcol * LDS_STRIDE + kA);
        v2f a  = {af.x, af.y};
        v2f bg = {bgf.x, bgf.y};
        v2f bu = {buf.x, buf.y};
        accg = wmma_f32_16x16x4(a, bg, accg);
        accu = wmma_f32_16x16x4(a, bu, accu);
      }
      cur ^= 1;
    }

    const int rowOff = (lane < 16) ? 0 : 8;
#pragma unroll
    for (int v = 0; v < 8; ++v) {
      const int m = mt * 16 + v + rowOff;
      if (m < nTok) {
        const float g = accg[v];
        const float u = accu[v];
        const float sg = g / (1.f + __expf(-g));   // SiLU
        mixed[(size_t)(base + m) * INTER_DIM + i0 + bcol] = sg * u;
      }
    }
    __syncthreads();  // all reads done before next mt's prologue DMA
  }
}

// ---------------- Phase 3: down projection per expert ------------------------
__global__ __launch_bounds__(256)
void moe_down_kernel(const float* __restrict__ mixed,
                     const float* __restrict__ wd,
                     const int* __restrict__ cnt,
                     const int* __restrict__ off,
                     float* __restrict__ out_pair) {
  __shared__ float lds_d[2][TILE_FLOATS];
  const int e = blockIdx.y;
  const int nTok = cnt[e];
  if (nTok == 0) return;
  const int base = off[e];
  const int h0t = blockIdx.x * 128;
  const int tid = threadIdx.x;
  const int lane = tid & 31;
  const int wave = tid >> 5;
  const int lhalf = (lane < 16) ? 0 : 2;
  const int l15 = lane & 15;
  const int bcol = wave * 16 + l15;

  const size_t wbase = ((size_t)e * HID + h0t) * INTER_DIM;
  const unsigned ldsd0 = (unsigned)(size_t)(void*)&lds_d[0][0];
  const unsigned ldsd1 = (unsigned)(size_t)(void*)&lds_d[1][0];

  const int nMT = (nTok + 15) >> 4;
  for (int mt = 0; mt < nMT; ++mt) {
    const int arow = mt * 16 + l15;
    const int pairRow = base + ((arow < nTok) ? arow : (nTok - 1));
    const float* mrow = mixed + (size_t)pairRow * INTER_DIM;

    v8f acc = {0.f, 0.f, 0.f, 0.f, 0.f, 0.f, 0.f, 0.f};

    if (wave == 0) tdm_load_tile_f32(ldsd0, wd + wbase, INTER_DIM);
    int cur = 0;
    for (int i0 = 0; i0 < INTER_DIM; i0 += BK) {
      if (wave == 0) __builtin_amdgcn_s_wait_tensorcnt(0);
      __syncthreads();
      if (wave == 0 && (i0 + BK) < INTER_DIM)
        tdm_load_tile_f32(cur ? ldsd0 : ldsd1, wd + wbase + i0 + BK, INTER_DIM);
      const float* bd_t = &lds_d[cur][0];
#pragma unroll
      for (int kk = 0; kk < BK; kk += 4) {
        const int kA = kk + lhalf;
        const float2 af = *(const float2*)(mrow + i0 + kA);
        const float2 bf = *(const float2*)(bd_t + bcol * LDS_STRIDE + kA);
        v2f a = {af.x, af.y};
        v2f b = {bf.x, bf.y};
        acc = wmma_f32_16x16x4(a, b, acc);
      }
      cur ^= 1;
    }

    const int rowOff = (lane < 16) ? 0 : 8;
#pragma unroll
    for (int v = 0; v < 8; ++v) {
      const int m = mt * 16 + v + rowOff;
      if (m < nTok)
        out_pair[(size_t)(base + m) * HID + h0t + bcol] = acc[v];
    }
    __syncthreads();
  }
}

// ---------------- Phase 4: deterministic weighted reduce ---------------------
__global__ __launch_bounds__(256)
void moe_reduce_kernel(const float* __restrict__ out_pair,
                       const int* __restrict__ pairof,
                       const float* __restrict__ wk,
                       float* __restrict__ out) {
  const int t = blockIdx.y;
  const int h = blockIdx.x * 256 + threadIdx.x;
  float acc = 0.f;
#pragma unroll
  for (int k = 0; k < TOPK; ++k) {
    const int p = pairof[t * TOPK + k];
    acc += wk[t * TOPK + k] * out_pair[(size_t)p * HID + h];
  }
  out[(size_t)t * HID + h] = acc;
}

// -----------------------------------------------------------------------------
extern "C" void kernel_launch(void* const* d_in, const int* in_sizes, int n_in,
                              void* d_out, int out_size, void* d_ws, size_t ws_size,
                              hipStream_t stream) {
  (void)in_sizes; (void)n_in; (void)out_size; (void)ws_size;
  const float* x  = (const float*)d_in[0];
  const float* rl = (const float*)d_in[1];
  const float* wg = (const float*)d_in[2];
  const float* wu = (const float*)d_in[3];
  const float* wd = (const float*)d_in[4];
  float* out = (float*)d_out;

  char* ws = (char*)d_ws;
  size_t pos = 0;
  auto alloc = [&](size_t bytes) -> void* {
    void* r = ws + pos;
    pos += bytes;
    pos = (pos + 255) & ~(size_t)255;
    return r;
  };
  int*   cnt      = (int*)alloc(N_EXP * sizeof(int));
  int*   off      = (int*)alloc((N_EXP + 1) * sizeof(int));
  int*   ptok     = (int*)alloc(T_TOKENS * TOPK * sizeof(int));
  int*   pairof   = (int*)alloc(T_TOKENS * TOPK * sizeof(int));
  float* wk       = (float*)alloc(T_TOKENS * TOPK * sizeof(float));
  float* mixed    = (float*)alloc((size_t)T_TOKENS * TOPK * INTER_DIM * sizeof(float));
  float* out_pair = (float*)alloc((size_t)T_TOKENS * TOPK * HID * sizeof(float));

  moe_route_kernel<<<1, 64, 0, stream>>>(rl, cnt, off, ptok, pairof, wk);

  dim3 g2(INTER_DIM / 128, N_EXP);
  moe_gateup_kernel<<<g2, 256, 0, stream>>>(x, wg, wu, cnt, off, ptok, mixed);

  dim3 g3(HID / 128, N_EXP);
  moe_down_kernel<<<g3, 256, 0, stream>>>(mixed, wd, cnt, off, out_pair);

  dim3 g4(HID / 256, T_TOKENS);
  moe_reduce_kernel<<<g4, 256, 0, stream>>>(out_pair, pairof, wk, out);
}